// MuCoMiD_506806141439
// MI455X (gfx1250) — compile-verified
//
#include <hip/hip_runtime.h>
#include <math.h>

#define EMB 256
#define HID 128

typedef __attribute__((ext_vector_type(2))) float v2f;
typedef __attribute__((ext_vector_type(8))) float v8f;

// ---------------------------------------------------------------------------
// deg[i] = 1.0 (self-loop weight) then += w[e] for every edge with dst==i,
// then deg -> rsqrt(deg) in place (the "dinv" vector of the reference).
// ---------------------------------------------------------------------------
__global__ void k_init_deg(float* __restrict__ deg, int n) {
  int i = blockIdx.x * blockDim.x + threadIdx.x;
  if (i < n) deg[i] = 1.0f;
}

__global__ void k_accum_deg(const int* __restrict__ el, const float* __restrict__ w,
                            float* __restrict__ deg, int ne) {
  int e = blockIdx.x * blockDim.x + threadIdx.x;
  if (e < ne) atomicAdd(&deg[el[2 * e + 1]], w[e]);
}

__global__ void k_rsqrt(float* __restrict__ deg, int n) {
  int i = blockIdx.x * blockDim.x + threadIdx.x;
  if (i < n) {
    float d = deg[i];
    deg[i] = d > 0.0f ? rsqrtf(d) : 0.0f;
  }
}

__global__ void k_zero(float* __restrict__ p, long long n) {
  long long i = (long long)blockIdx.x * blockDim.x + threadIdx.x;
  if (i < n) p[i] = 0.0f;
}

// ---------------------------------------------------------------------------
// xw[n x 128] = x[n x 256] @ W[256 x 128]  via V_WMMA_F32_16X16X4_F32.
// One wave32 computes one 16x16 tile; block = 256 threads = 8 waves covers
// the full N=128 for one 16-row block. K=256 -> 64 chained WMMAs.
//
// fp32 A (16x4) layout (ISA 7.12.2): lanes 0-15 row M=lane, VGPR0=K0/VGPR1=K1;
// lanes 16-31 row M=lane-16, VGPR0=K2/VGPR1=K3. B (4x16) analogous with N
// striped across lanes. C/D: VGPR r holds (M=r, N=lane) lanes 0-15 and
// (M=r+8, N=lane-16) lanes 16-31.
// n is a multiple of 16 here (30000, 60000) so EXEC is all-ones for WMMA.
// ---------------------------------------------------------------------------
__global__ void k_gemm_wmma(const float* __restrict__ x, const float* __restrict__ W,
                            float* __restrict__ xw, int n) {
  const int wave = threadIdx.x >> 5;       // 0..7 -> N tile index
  const int lane = threadIdx.x & 31;
  const int half = lane >> 4;              // 0: lanes 0-15, 1: lanes 16-31
  const int l    = lane & 15;
  const int m0   = blockIdx.x * 16;
  const int n0   = wave * 16;

  v8f c = {};
  const float* arow = x + (size_t)(m0 + l) * EMB + 2 * half;   // A row l, K-phase
  const float* bcol = W + (size_t)(2 * half) * HID + n0 + l;   // B K-phase row, col l

  for (int k = 0; k < EMB; k += 4) {
    v2f a, b;
    a.x = arow[k + 0];                     // K = k + 2*half
    a.y = arow[k + 1];                     // K = k + 2*half + 1
    b.x = bcol[(size_t)k * HID];           // K = k + 2*half
    b.y = bcol[(size_t)(k + 1) * HID];     // K = k + 2*half + 1
    c = __builtin_amdgcn_wmma_f32_16x16x4_f32(
        /*neg_a=*/false, a, /*neg_b=*/false, b,
        /*c_mod=*/(short)0, c, /*reuse_a=*/false, /*reuse_b=*/false);
  }

  float* orow = xw + (size_t)(m0 + half * 8) * HID + n0 + l;
#pragma unroll
  for (int r = 0; r < 8; ++r) orow[(size_t)r * HID] = c[r];
}

// Scalar tail for rows not covered by 16-row WMMA tiles (not used for these
// sizes, but keeps the kernel correct for arbitrary n).
__global__ void k_gemm_tail(const float* __restrict__ x, const float* __restrict__ W,
                            float* __restrict__ xw, int row0, int n) {
  int gid = blockIdx.x * blockDim.x + threadIdx.x;
  int row = row0 + (gid >> 7);
  int col = gid & (HID - 1);
  if (row >= n) return;
  float acc = 0.0f;
  const float* xr = x + (size_t)row * EMB;
  for (int k = 0; k < EMB; ++k) acc = fmaf(xr[k], W[(size_t)k * HID + col], acc);
  xw[(size_t)row * HID + col] = acc;
}

// ---------------------------------------------------------------------------
// Edge scatter: acc[dst] += dinv[src]*w*dinv[dst] * xw[src]
// One wave32 per edge, 4 contiguous features per lane (float4 load + 4 atomics).
// ---------------------------------------------------------------------------
__global__ void k_scatter(const int* __restrict__ el, const float* __restrict__ w,
                          const float* __restrict__ dinv, const float* __restrict__ xw,
                          float* __restrict__ acc, int ne) {
  long long gid = (long long)blockIdx.x * blockDim.x + threadIdx.x;
  int e    = (int)(gid >> 5);
  int lane = (int)(gid & 31);
  if (e >= ne) return;
  int src = el[2 * e + 0];
  int dst = el[2 * e + 1];
  float nrm = dinv[src] * w[e] * dinv[dst];
  const float4 xv = *(const float4*)(xw + (size_t)src * HID + lane * 4);
  float* ap = acc + (size_t)dst * HID + lane * 4;
  atomicAdd(ap + 0, nrm * xv.x);
  atomicAdd(ap + 1, nrm * xv.y);
  atomicAdd(ap + 2, nrm * xv.z);
  atomicAdd(ap + 3, nrm * xv.w);
}

// hid = relu(acc + dinv^2 * xw + b)  (self-loop term folded in; in-place on acc)
__global__ void k_finalize(float* __restrict__ acc, const float* __restrict__ xw,
                           const float* __restrict__ dinv, const float* __restrict__ b,
                           int n) {
  long long gid = (long long)blockIdx.x * blockDim.x + threadIdx.x;
  int i = (int)(gid >> 7);
  int f = (int)(gid & (HID - 1));
  if (i >= n) return;
  float di = dinv[i];
  size_t off = (size_t)i * HID + f;
  float v = acc[off] + di * di * xw[off] + b[f];
  acc[off] = v > 0.0f ? v : 0.0f;
}

// ---------------------------------------------------------------------------
// Pair head: out[p] = sigmoid( sum_f ha[ia][f]*hb[ib][f]*Wv[f] + bv )
// One wave32 per pair, float4 per lane, wave shuffle reduction.
// ---------------------------------------------------------------------------
__global__ void k_pairs(const int* __restrict__ pairs, const float* __restrict__ ha,
                        const float* __restrict__ hb, const float* __restrict__ Wv,
                        const float* __restrict__ bv, float* __restrict__ out, int np) {
  long long gid = (long long)blockIdx.x * blockDim.x + threadIdx.x;
  int p    = (int)(gid >> 5);
  int lane = (int)(gid & 31);
  if (p >= np) return;
  int ia = pairs[2 * p + 0];
  int ib = pairs[2 * p + 1];
  const float4 a = *(const float4*)(ha + (size_t)ia * HID + lane * 4);
  const float4 b = *(const float4*)(hb + (size_t)ib * HID + lane * 4);
  const float4 w = *(const float4*)(Wv + lane * 4);
  float s = a.x * b.x * w.x + a.y * b.y * w.y + a.z * b.z * w.z + a.w * b.w * w.w;
#pragma unroll
  for (int off = 16; off > 0; off >>= 1) s += __shfl_down(s, off, 32);
  if (lane == 0) out[p] = 1.0f / (1.0f + __expf(-(s + bv[0])));
}

// ---------------------------------------------------------------------------
static void run_gcn(const float* x, const int* el, const float* w,
                    const float* W, const float* b,
                    float* xw, float* acc, float* deg,
                    int n, int ne, hipStream_t stream) {
  k_init_deg<<<(n + 255) / 256, 256, 0, stream>>>(deg, n);
  k_accum_deg<<<(ne + 255) / 256, 256, 0, stream>>>(el, w, deg, ne);
  k_rsqrt<<<(n + 255) / 256, 256, 0, stream>>>(deg, n);

  int nt = n / 16;                                   // full 16-row WMMA tiles
  if (nt > 0) k_gemm_wmma<<<nt, 256, 0, stream>>>(x, W, xw, n);
  int row0 = nt * 16;
  if (row0 < n) {
    long long t = (long long)(n - row0) * HID;
    k_gemm_tail<<<(int)((t + 255) / 256), 256, 0, stream>>>(x, W, xw, row0, n);
  }

  long long nelem = (long long)n * HID;
  k_zero<<<(int)((nelem + 255) / 256), 256, 0, stream>>>(acc, nelem);
  long long st = (long long)ne * 32;
  k_scatter<<<(int)((st + 255) / 256), 256, 0, stream>>>(el, w, deg, xw, acc, ne);
  k_finalize<<<(int)((nelem + 255) / 256), 256, 0, stream>>>(acc, xw, deg, b, n);
}

extern "C" void kernel_launch(void* const* d_in, const int* in_sizes, int n_in,
                              void* d_out, int out_size, void* d_ws, size_t ws_size,
                              hipStream_t stream) {
  const float* memb = (const float*)d_in[0];
  const float* demb = (const float*)d_in[1];
  const float* pemb = (const float*)d_in[2];
  const int*   m_el = (const int*)d_in[3];
  const float* m_w  = (const float*)d_in[4];
  const int*   d_el = (const int*)d_in[5];
  const float* d_w  = (const float*)d_in[6];
  const int*   p_el = (const int*)d_in[7];
  const float* p_w  = (const float*)d_in[8];
  const int* mp_pairs = (const int*)d_in[9];
  const int* dp_pairs = (const int*)d_in[10];
  const int* md_pairs = (const int*)d_in[11];
  const float* Wm = (const float*)d_in[12];
  const float* bm = (const float*)d_in[13];
  const float* Wd = (const float*)d_in[14];
  const float* bd = (const float*)d_in[15];
  const float* Wp = (const float*)d_in[16];
  const float* bp = (const float*)d_in[17];
  const float* W_assoc = (const float*)d_in[18];
  const float* b_assoc = (const float*)d_in[19];
  const float* W_mp = (const float*)d_in[20];
  const float* b_mp = (const float*)d_in[21];
  const float* W_dp = (const float*)d_in[22];
  const float* b_dp = (const float*)d_in[23];

  const int NM = in_sizes[0] / EMB;     // 30000
  const int ND = in_sizes[1] / EMB;     // 30000
  const int NP = in_sizes[2] / EMB;     // 60000
  const int EM = in_sizes[4];           // 480000
  const int ED = in_sizes[6];           // 480000
  const int EP = in_sizes[8];           // 960000
  const int NPAIR = in_sizes[9] / 2;    // 250000

  // Workspace partition (~124 MB of floats)
  float* ws = (float*)d_ws;
  float* xw_m  = ws; ws += (size_t)NM * HID;
  float* xw_d  = ws; ws += (size_t)ND * HID;
  float* xw_p  = ws; ws += (size_t)NP * HID;
  float* acc_m = ws; ws += (size_t)NM * HID;   // becomes mhid after finalize
  float* acc_d = ws; ws += (size_t)ND * HID;   // dhid
  float* acc_p = ws; ws += (size_t)NP * HID;   // phid
  float* deg_m = ws; ws += NM;                 // becomes dinv in place
  float* deg_d = ws; ws += ND;
  float* deg_p = ws; ws += NP;

  run_gcn(memb, m_el, m_w, Wm, bm, xw_m, acc_m, deg_m, NM, EM, stream);
  run_gcn(demb, d_el, d_w, Wd, bd, xw_d, acc_d, deg_d, ND, ED, stream);
  run_gcn(pemb, p_el, p_w, Wp, bp, xw_p, acc_p, deg_p, NP, EP, stream);

  float* out = (float*)d_out;
  long long pt = (long long)NPAIR * 32;
  int pblocks = (int)((pt + 255) / 256);
  // return order: (assoc, mp, dp)
  k_pairs<<<pblocks, 256, 0, stream>>>(md_pairs, acc_m, acc_d, W_assoc, b_assoc,
                                       out, NPAIR);
  k_pairs<<<pblocks, 256, 0, stream>>>(mp_pairs, acc_m, acc_p, W_mp, b_mp,
                                       out + NPAIR, NPAIR);
  k_pairs<<<pblocks, 256, 0, stream>>>(dp_pairs, acc_d, acc_p, W_dp, b_dp,
                                       out + 2 * (size_t)NPAIR, NPAIR);
}